// GCN_64080912056900
// MI455X (gfx1250) — compile-verified
//
#include <hip/hip_runtime.h>

typedef float v2f __attribute__((ext_vector_type(2)));
typedef float v8f __attribute__((ext_vector_type(8)));
typedef unsigned int u32x4 __attribute__((ext_vector_type(4)));
typedef int i32x4 __attribute__((ext_vector_type(4)));
typedef int i32x8 __attribute__((ext_vector_type(8)));

#define KIN 512
#define F1  16
#define F2  40
#define F2P 48   // 40 padded to 3 WMMA N-tiles of 16

#if defined(__has_builtin)
#if __has_builtin(__builtin_amdgcn_tensor_load_to_lds) && __has_builtin(__builtin_amdgcn_s_wait_tensorcnt)
#define HAVE_TDM 1
#endif
#endif

// ---------------- degree / norm precompute ----------------

__global__ __launch_bounds__(256) void k_deg(const int* __restrict__ row,
                                             float* __restrict__ deg, int E) {
  int e = blockIdx.x * 256 + threadIdx.x;
  if (e < E) atomicAdd(&deg[row[e]], 1.0f);
}

__global__ __launch_bounds__(256) void k_dinv(float* __restrict__ deg, int n) {
  int i = blockIdx.x * 256 + threadIdx.x;
  if (i < n) deg[i] = rsqrtf(deg[i] + 1.0f);  // +1 self-loop => deg > 0 always
}

__global__ __launch_bounds__(256) void k_enorm(const int* __restrict__ row,
                                               const int* __restrict__ col,
                                               const float* __restrict__ dinv,
                                               float* __restrict__ enorm, int E) {
  int e = blockIdx.x * 256 + threadIdx.x;
  if (e < E) enorm[e] = dinv[row[e]] * dinv[col[e]];
}

// ---------------- TDM helper: stage `nfloats` contiguous floats into LDS ----------------
// D# packing per CDNA5 ISA ch.8 (§8.3/8.4): 1-D tensor viewed as a 1-row 2-D tile.

__device__ __forceinline__ void tdm_stage_to_lds(const float* gsrc, void* lds_dst,
                                                 unsigned nfloats) {
#ifdef HAVE_TDM
  unsigned long long ga = (unsigned long long)(uintptr_t)gsrc;
  unsigned la = (unsigned)(uintptr_t)lds_dst;  // flat shared addr: low 32 bits = LDS offset

  u32x4 g0;
  g0.x = 0x1u;                                   // count=1, is_restore=0, gather off
  g0.y = la;                                     // lds_addr
  g0.z = (unsigned)(ga & 0xFFFFFFFFu);           // global_addr[31:0]
  g0.w = (unsigned)((ga >> 32) & 0x1FFFFFFu)     // global_addr[56:32]
         | (2u << 30);                           // type=2 ("image")

  i32x8 g1;
  g1[0] = (int)(2u << 16);                       // data_size=2 (4 bytes); wg_mask=0; no flags
  g1[1] = (int)((nfloats & 0xFFFFu) << 16);      // tensor_dim0[15:0]  @ bits[63:48]
  g1[2] = (int)((nfloats >> 16) | (1u << 16));   // tensor_dim0[31:16] | tensor_dim1=1
  g1[3] = (int)((nfloats & 0xFFFFu) << 16);      // tile_dim0 @ bits[127:112]
  g1[4] = 1;                                     // tile_dim1=1, tile_dim2=0
  g1[5] = (int)nfloats;                          // tensor_dim0_stride[31:0]
  g1[6] = 0;                                     // stride hi / dim1_stride lo
  g1[7] = 0;

  i32x4 gz = {0, 0, 0, 0};                       // groups 2/3 unused (<=2-D)
#if __clang_major__ >= 23
  i32x8 gz8 = {0, 0, 0, 0, 0, 0, 0, 0};
  __builtin_amdgcn_tensor_load_to_lds(g0, g1, gz, gz, gz8, 0);
#else
  __builtin_amdgcn_tensor_load_to_lds(g0, g1, gz, gz, 0);
#endif
  __builtin_amdgcn_s_wait_tensorcnt(0);
#else
  (void)gsrc; (void)lds_dst; (void)nfloats;
#endif
}

// ---------------- layer-1 GEMM: h[N,16] = x[N,512] @ W1[512,16] ----------------
// one wave32 per 16-row tile; V_WMMA_F32_16X16X4_F32, K-loop of 128 steps.
// A 16x4 f32: lane(0..15)->M=lane K={0,1}; lane(16..31)->M=lane-16 K={2,3}
// B 4x16 f32: lane(0..15)->N=lane K={0,1}; lane(16..31)->N=lane-16 K={2,3}
// C/D 16x16:  vgpr r holds M = r + 8*(lane>=16), N = lane&15

__global__ __launch_bounds__(256) void k_gemm1(const float* __restrict__ x,
                                               const float* __restrict__ W1,
                                               float* __restrict__ h,
                                               int nrows, int ntiles) {
  __shared__ float w1s[KIN * F1];  // 32 KB
#ifdef HAVE_TDM
  if (threadIdx.x < 32) tdm_stage_to_lds(W1, w1s, KIN * F1);  // wave0 issues TDM DMA
#else
  for (int i = threadIdx.x; i < KIN * F1; i += 256) w1s[i] = W1[i];
#endif
  __syncthreads();

  int wave = threadIdx.x >> 5, lane = threadIdx.x & 31;
  int tile = blockIdx.x * 8 + wave;
  if (tile >= ntiles) return;  // whole-wave exit: EXEC stays all-1s for WMMA

  int hi = lane >> 4;      // K-half select
  int n  = lane & 15;      // row-in-tile for A, column for B/D
  int arow = tile * 16 + n; if (arow >= nrows) arow = nrows - 1;
  const float* xr = x + (size_t)arow * KIN + 2 * hi;

  v8f acc = {0.f, 0.f, 0.f, 0.f, 0.f, 0.f, 0.f, 0.f};
#pragma unroll 8
  for (int k0 = 0; k0 < KIN; k0 += 4) {
    v2f a, b;
    a.x = xr[k0];
    a.y = xr[k0 + 1];
    int kb = k0 + 2 * hi;
    b.x = w1s[kb * F1 + n];
    b.y = w1s[kb * F1 + F1 + n];
    acc = __builtin_amdgcn_wmma_f32_16x16x4_f32(false, a, false, b,
                                                (short)0, acc, false, false);
  }
  if (tile * 16 + 16 <= nrows) {  // uniform fast path: full tile, unguarded stores
    float* hb = h + ((size_t)(tile * 16) + 8 * hi) * F1 + n;
#pragma unroll
    for (int r = 0; r < 8; ++r) hb[(size_t)r * F1] = acc[r];
  } else {
#pragma unroll
    for (int r = 0; r < 8; ++r) {
      int m = tile * 16 + r + 8 * hi;
      if (m < nrows) h[(size_t)m * F1 + n] = acc[r];
    }
  }
}

// ---------------- layer-1 edge scatter: agg1[row] += h[col] * norm ----------------

__global__ __launch_bounds__(256) void k_scatter1(const int* __restrict__ row,
                                                  const int* __restrict__ col,
                                                  const float* __restrict__ enorm,
                                                  const float* __restrict__ h,
                                                  float* __restrict__ agg,
                                                  long long total) {
  long long idx = (long long)blockIdx.x * 256 + threadIdx.x;
  if (idx >= total) return;
  int e = (int)(idx >> 4), f = (int)(idx & 15);
  atomicAdd(&agg[(size_t)row[e] * F1 + f], h[(size_t)col[e] * F1 + f] * enorm[e]);
}

// ---------------- layer-1 finish: self-loop + bias + ReLU (in place) ----------------

__global__ __launch_bounds__(256) void k_finish1(float* __restrict__ agg1,
                                                 const float* __restrict__ h,
                                                 const float* __restrict__ dinv,
                                                 const float* __restrict__ b1,
                                                 int total) {
  int idx = blockIdx.x * 256 + threadIdx.x;
  if (idx >= total) return;
  int i = idx >> 4, f = idx & 15;
  float d = dinv[i];
  float v = agg1[idx] + h[idx] * d * d + b1[f];
  agg1[idx] = fmaxf(v, 0.0f);
}

// ---------------- layer-2 GEMM: h2[N,40] = h1[N,16] @ W2[16,40] ----------------

__global__ __launch_bounds__(256) void k_gemm2(const float* __restrict__ h1,
                                               const float* __restrict__ W2,
                                               float* __restrict__ h2,
                                               int nrows, int ntiles) {
  __shared__ float w2s[F1 * F2P];  // 16x48, zero padded (3 KB; manual copy)
  for (int i = threadIdx.x; i < F1 * F2P; i += 256) {
    int k = i / F2P, nn = i % F2P;
    w2s[i] = (nn < F2) ? W2[k * F2 + nn] : 0.0f;
  }
  __syncthreads();

  int wave = threadIdx.x >> 5, lane = threadIdx.x & 31;
  int tile = blockIdx.x * 8 + wave;
  if (tile >= ntiles) return;

  int hi = lane >> 4, n = lane & 15;
  int arow = tile * 16 + n; if (arow >= nrows) arow = nrows - 1;
  const float* ar = h1 + (size_t)arow * F1 + 2 * hi;

  v8f c0 = {0.f,0.f,0.f,0.f,0.f,0.f,0.f,0.f};
  v8f c1 = c0, c2 = c0;
#pragma unroll
  for (int k0 = 0; k0 < F1; k0 += 4) {
    v2f a;
    a.x = ar[k0];
    a.y = ar[k0 + 1];
    int kb = k0 + 2 * hi;
    v2f b0, b1v, b2v;
    b0.x  = w2s[kb * F2P + n];       b0.y  = w2s[kb * F2P + F2P + n];
    b1v.x = w2s[kb * F2P + 16 + n];  b1v.y = w2s[kb * F2P + F2P + 16 + n];
    b2v.x = w2s[kb * F2P + 32 + n];  b2v.y = w2s[kb * F2P + F2P + 32 + n];
    c0 = __builtin_amdgcn_wmma_f32_16x16x4_f32(false, a, false, b0,  (short)0, c0, false, false);
    c1 = __builtin_amdgcn_wmma_f32_16x16x4_f32(false, a, false, b1v, (short)0, c1, false, false);
    c2 = __builtin_amdgcn_wmma_f32_16x16x4_f32(false, a, false, b2v, (short)0, c2, false, false);
  }
  if (tile * 16 + 16 <= nrows) {  // uniform fast path
    size_t base = (size_t)(tile * 16 + 8 * hi) * F2;
#pragma unroll
    for (int r = 0; r < 8; ++r) {
      h2[base + (size_t)r * F2 + n]      = c0[r];
      h2[base + (size_t)r * F2 + 16 + n] = c1[r];
      if (32 + n < F2) h2[base + (size_t)r * F2 + 32 + n] = c2[r];
    }
  } else {
#pragma unroll
    for (int r = 0; r < 8; ++r) {
      int m = tile * 16 + r + 8 * hi;
      if (m < nrows) {
        size_t base = (size_t)m * F2;
        h2[base + n]      = c0[r];
        h2[base + 16 + n] = c1[r];
        if (32 + n < F2) h2[base + 32 + n] = c2[r];
      }
    }
  }
}

// ---------------- layer-2 edge scatter ----------------

__global__ __launch_bounds__(256) void k_scatter2(const int* __restrict__ row,
                                                  const int* __restrict__ col,
                                                  const float* __restrict__ enorm,
                                                  const float* __restrict__ h2,
                                                  float* __restrict__ agg,
                                                  long long total) {
  long long idx = (long long)blockIdx.x * 256 + threadIdx.x;
  if (idx >= total) return;
  int e = (int)(idx / F2);
  int f = (int)(idx - (long long)e * F2);
  atomicAdd(&agg[(size_t)row[e] * F2 + f], h2[(size_t)col[e] * F2 + f] * enorm[e]);
}

// ---------------- final: self-loop + bias + log_softmax, one wave32 per node ----------------

__global__ __launch_bounds__(256) void k_final(const float* __restrict__ agg2,
                                               const float* __restrict__ h2,
                                               const float* __restrict__ dinv,
                                               const float* __restrict__ b2,
                                               float* __restrict__ out, int n) {
  int wave = threadIdx.x >> 5, lane = threadIdx.x & 31;
  int i = blockIdx.x * 8 + wave;
  if (i >= n) return;
  float d = dinv[i];
  float d2 = d * d;
  size_t base = (size_t)i * F2;
  float v0 = agg2[base + lane] + h2[base + lane] * d2 + b2[lane];           // f = 0..31
  float v1 = (lane < F2 - 32)
                 ? agg2[base + 32 + lane] + h2[base + 32 + lane] * d2 + b2[32 + lane]
                 : -__builtin_inff();                                        // f = 32..39
  float m = fmaxf(v0, v1);
#pragma unroll
  for (int o = 16; o > 0; o >>= 1) m = fmaxf(m, __shfl_xor(m, o));
  float s = __expf(v0 - m) + ((lane < F2 - 32) ? __expf(v1 - m) : 0.0f);
#pragma unroll
  for (int o = 16; o > 0; o >>= 1) s += __shfl_xor(s, o);
  float lse = m + __logf(s);
  out[base + lane] = v0 - lse;
  if (lane < F2 - 32) out[base + 32 + lane] = v1 - lse;
}

// ---------------- host ----------------

extern "C" void kernel_launch(void* const* d_in, const int* in_sizes, int n_in,
                              void* d_out, int out_size, void* d_ws, size_t ws_size,
                              hipStream_t stream) {
  const float* x  = (const float*)d_in[0];
  const int*   ei = (const int*)d_in[1];   // edge_index [2,E] (int32 per harness)
  const float* W1 = (const float*)d_in[2];
  const float* b1 = (const float*)d_in[3];
  const float* W2 = (const float*)d_in[4];
  const float* b2 = (const float*)d_in[5];

  const int n = in_sizes[0] / KIN;
  const int E = in_sizes[1] / 2;
  const int* row = ei;
  const int* col = ei + E;

  // workspace carve-out (floats): dinv | h | agg1 | h2 | agg2 | enorm  (~58 MB)
  float* ws    = (float*)d_ws;
  float* dinv  = ws;
  float* h     = dinv + (((size_t)n + 127) & ~(size_t)127);
  float* agg1  = h    + (size_t)n * F1;
  float* h2    = agg1 + (size_t)n * F1;
  float* agg2  = h2   + (size_t)n * F2;
  float* enorm = agg2 + (size_t)n * F2;

  hipMemsetAsync(dinv, 0, (size_t)n * sizeof(float), stream);
  hipMemsetAsync(agg1, 0, (size_t)n * F1 * sizeof(float), stream);
  hipMemsetAsync(agg2, 0, (size_t)n * F2 * sizeof(float), stream);

  k_deg  <<<(E + 255) / 256, 256, 0, stream>>>(row, dinv, E);
  k_dinv <<<(n + 255) / 256, 256, 0, stream>>>(dinv, n);
  k_enorm<<<(E + 255) / 256, 256, 0, stream>>>(row, col, dinv, enorm, E);

  const int ntiles = (n + 15) / 16;
  k_gemm1<<<(ntiles + 7) / 8, 256, 0, stream>>>(x, W1, h, n, ntiles);

  const long long t1 = (long long)E * F1;
  k_scatter1<<<(unsigned)((t1 + 255) / 256), 256, 0, stream>>>(row, col, enorm, h, agg1, t1);

  const int tot1 = n * F1;
  k_finish1<<<(tot1 + 255) / 256, 256, 0, stream>>>(agg1, h, dinv, b1, tot1);

  k_gemm2<<<(ntiles + 7) / 8, 256, 0, stream>>>(agg1, W2, h2, n, ntiles);

  const long long t2 = (long long)E * F2;
  k_scatter2<<<(unsigned)((t2 + 255) / 256), 256, 0, stream>>>(row, col, enorm, h2, agg2, t2);

  k_final<<<(n + 7) / 8, 256, 0, stream>>>(agg2, h2, dinv, b2, (float*)d_out, n);
}